// EncoderRNN_27711128994599
// MI455X (gfx1250) — compile-verified
//
#include <hip/hip_runtime.h>
#include <hip/hip_bf16.h>
#include <math.h>

// Problem constants (E,H,V,L,B,S) = (512, 1024, 32000, 2, 128, 64)
#define EE   512
#define HH   1024
#define VV   32000
#define BB   128
#define SS   64

typedef float v2f __attribute__((ext_vector_type(2)));
typedef float v8f __attribute__((ext_vector_type(8)));

__device__ __forceinline__ v8f wmma_f32(v2f a, v2f b, v8f c) {
  // D = A(16x4) * B(4x16) + C(16x16), fp32 CDNA5 WMMA
  return __builtin_amdgcn_wmma_f32_16x16x4_f32(false, a, false, b, (short)0, c,
                                               false, false);
}

__device__ __forceinline__ float sigmoidf_(float x) {
  return 1.0f / (1.0f + __expf(-x));
}

// ---------------------------------------------------------------------------
// meta[t] (t=0..64) = packed-row offset = sum_b min(len_b, t)
// meta[65] = bs_last (count of len == maxlen), meta[66] = P = sum(len)
// ---------------------------------------------------------------------------
__global__ void meta_kernel(const int* __restrict__ lengths, int* __restrict__ meta) {
  int t = threadIdx.x;
  if (t <= 64) {
    int s = 0;
    for (int b = 0; b < BB; ++b) {
      int l = lengths[b];
      s += (l < t) ? l : t;
    }
    meta[t] = s;
  }
  if (t == 65) {
    int maxlen = lengths[0];  // lengths sorted descending
    int cnt = 0, P = 0;
    for (int b = 0; b < BB; ++b) {
      cnt += (lengths[b] == maxlen);
      P += lengths[b];
    }
    meta[65] = cnt;
    meta[66] = P;
  }
}

__global__ void zero_kernel(float* __restrict__ p, int n) {
  int i = blockIdx.x * blockDim.x + threadIdx.x;
  if (i < n) p[i] = 0.0f;
}

// ---------------------------------------------------------------------------
// One fused LSTM step: gates = [x|h] @ [w_ih|w_hh]^T + b, then i/f/g/o
// nonlinearity + masked c/h update, all in registers.
// Block = 128 threads (4 waves). Wave wv -> 32 b-rows [32*wv, 32*wv+32).
// blockIdx.x -> 16 h-cols [16*bx, 16*bx+16). Grid = 64 blocks.
// Per k-step: 2 A fragments (two 16-row tiles) x 4 gate B fragments
// -> 8 independent WMMA streams; weight fragments reused across both A tiles.
// ---------------------------------------------------------------------------
__global__ __launch_bounds__(128)
void lstm_step_kernel(const float* __restrict__ x_feat,   // [B][E]   (layer0, t==0)
                      const int*   __restrict__ tokens,   // [B][S]   (layer0, t>0)
                      const float* __restrict__ embed_W,  // [V][E]
                      const float* __restrict__ x_seq,    // layer1: hs0[t] base, else null
                      const float* __restrict__ h_in,     // [B][H]
                      float*       __restrict__ h_out,    // [B][H]
                      float*       __restrict__ c_state,  // [B][H]
                      float*       __restrict__ hs_out,   // hs[t] base [B][H]
                      const float* __restrict__ w_ih,     // [4H][Kx]
                      const float* __restrict__ w_hh,     // [4H][H]
                      const float* __restrict__ b_ih,     // [4H]
                      const float* __restrict__ b_hh,     // [4H]
                      const int*   __restrict__ lengths,  // [B]
                      int t, int Kx) {
  const int lane = threadIdx.x & 31;
  const int wv   = threadIdx.x >> 5;   // 0..3 : 32-row b-tile
  const int lrow = lane & 15;
  const int lhi  = lane >> 4;          // 0/1
  const int h0   = blockIdx.x * 16;
  const int b0   = wv * 32;

  const int hcol = h0 + lrow;          // B-matrix row (output col) this lane streams

  // Two A-tile row pointers per lane (rows b0+lrow and b0+16+lrow)
  const float* xrow[2];
  const float* hrow[2];
#pragma unroll
  for (int m = 0; m < 2; ++m) {
    int brow = b0 + 16 * m + lrow;
    if (x_seq)       xrow[m] = x_seq   + (size_t)brow * Kx;
    else if (t == 0) xrow[m] = x_feat  + (size_t)brow * EE;
    else             xrow[m] = embed_W + (size_t)tokens[brow * SS + (t - 1)] * EE;
    hrow[m] = h_in + (size_t)brow * HH;
  }

  const float* wihr[4];
  const float* whhr[4];
#pragma unroll
  for (int g = 0; g < 4; ++g) {
    int n = g * HH + hcol;
    wihr[g] = w_ih + (size_t)n * Kx;
    whhr[g] = w_hh + (size_t)n * HH;
  }

  v8f acc[2][4] = {};
  const int k2 = 2 * lhi;

  // x @ w_ih^T contribution
#pragma unroll 2
  for (int k = 0; k < Kx; k += 4) {
    v2f a0 = *(const v2f*)(xrow[0] + k + k2);
    v2f a1 = *(const v2f*)(xrow[1] + k + k2);
#pragma unroll
    for (int g = 0; g < 4; ++g) {
      v2f bf = *(const v2f*)(wihr[g] + k + k2);
      acc[0][g] = wmma_f32(a0, bf, acc[0][g]);
      acc[1][g] = wmma_f32(a1, bf, acc[1][g]);
    }
  }
  // h @ w_hh^T contribution
#pragma unroll 2
  for (int k = 0; k < HH; k += 4) {
    v2f a0 = *(const v2f*)(hrow[0] + k + k2);
    v2f a1 = *(const v2f*)(hrow[1] + k + k2);
#pragma unroll
    for (int g = 0; g < 4; ++g) {
      v2f bf = *(const v2f*)(whhr[g] + k + k2);
      acc[0][g] = wmma_f32(a0, bf, acc[0][g]);
      acc[1][g] = wmma_f32(a1, bf, acc[1][g]);
    }
  }

  float bias[4];
#pragma unroll
  for (int g = 0; g < 4; ++g) bias[g] = b_ih[g * HH + hcol] + b_hh[g * HH + hcol];

  // C/D layout: element (r, lane) -> M = r + 8*lhi, N = lrow
#pragma unroll
  for (int m = 0; m < 2; ++m) {
#pragma unroll
    for (int r = 0; r < 8; ++r) {
      int b   = b0 + 16 * m + r + 8 * lhi;
      int idx = b * HH + hcol;
      float iv = sigmoidf_(acc[m][0][r] + bias[0]);
      float fv = sigmoidf_(acc[m][1][r] + bias[1]);
      float gv = tanhf    (acc[m][2][r] + bias[2]);
      float ov = sigmoidf_(acc[m][3][r] + bias[3]);
      float c_old = c_state[idx];
      float h_old = h_in[idx];
      float c_new = fv * c_old + iv * gv;
      float h_new = ov * tanhf(c_new);
      bool  act   = t < lengths[b];
      float hs = act ? h_new : h_old;
      float cs = act ? c_new : c_old;
      c_state[idx] = cs;
      h_out[idx]   = hs;
      hs_out[idx]  = hs;
    }
  }
}

// ---------------------------------------------------------------------------
// Packed projection: out[offset[t]+b][v] = hs1[t][b] . lin_W[v] + lin_b[v]
// Block = 128 threads (4 waves); wave -> 32 rows x 64 V-cols; block -> 32x256.
// blockIdx.x in [0,125) covers V=32000; blockIdx.y in [0,256) = (t, 32-row tile).
// lin_W fragments reused across both 16-row A tiles.
// Uniform early-exit for fully inactive tiles (lengths sorted descending).
// ---------------------------------------------------------------------------
__global__ __launch_bounds__(128)
void proj_kernel(const float* __restrict__ hs1,
                 const float* __restrict__ lin_W,   // [V][H]
                 const float* __restrict__ lin_b,   // [V]
                 const int*   __restrict__ lengths,
                 const int*   __restrict__ meta,
                 float*       __restrict__ out) {
  const int lane = threadIdx.x & 31;
  const int wv   = threadIdx.x >> 5;   // 0..3
  const int lrow = lane & 15;
  const int lhi  = lane >> 4;
  const int rt   = blockIdx.y;         // 0..255
  const int t    = rt >> 2;            // 0..63
  const int b0   = (rt & 3) * 32;

  if (lengths[b0] <= t) return;  // block-uniform: all 32 rows inactive

  const int vbase = blockIdx.x * 256 + wv * 64;

  const float* arow[2];
#pragma unroll
  for (int m = 0; m < 2; ++m)
    arow[m] = hs1 + ((size_t)t * BB + b0 + 16 * m + lrow) * HH;

  const float* wrow[4];
#pragma unroll
  for (int j = 0; j < 4; ++j)
    wrow[j] = lin_W + (size_t)(vbase + j * 16 + lrow) * HH;

  v8f acc[2][4] = {};
  const int k2 = 2 * lhi;
#pragma unroll 2
  for (int k = 0; k < HH; k += 4) {
    v2f a0 = *(const v2f*)(arow[0] + k + k2);
    v2f a1 = *(const v2f*)(arow[1] + k + k2);
#pragma unroll
    for (int j = 0; j < 4; ++j) {
      v2f bf = *(const v2f*)(wrow[j] + k + k2);
      acc[0][j] = wmma_f32(a0, bf, acc[0][j]);
      acc[1][j] = wmma_f32(a1, bf, acc[1][j]);
    }
  }

  float lb[4];
#pragma unroll
  for (int j = 0; j < 4; ++j) lb[j] = lin_b[vbase + j * 16 + lrow];

  const int pbase = meta[t];
#pragma unroll
  for (int m = 0; m < 2; ++m) {
#pragma unroll
    for (int r = 0; r < 8; ++r) {
      int b = b0 + 16 * m + r + 8 * lhi;
      if (lengths[b] > t) {
        size_t row = (size_t)(pbase + b) * VV;
#pragma unroll
        for (int j = 0; j < 4; ++j)
          out[row + vbase + j * 16 + lrow] = acc[m][j][r] + lb[j];
      }
    }
  }
}

// ---------------------------------------------------------------------------
// Final h_t / c_t: stack layer0/layer1 finals for b < bs_last after outputs.
// ---------------------------------------------------------------------------
__global__ void finals_kernel(const float* __restrict__ h0f,
                              const float* __restrict__ h1f,
                              const float* __restrict__ c0f,
                              const float* __restrict__ c1f,
                              const int*   __restrict__ meta,
                              float*       __restrict__ out) {
  int i = blockIdx.x * blockDim.x + threadIdx.x;  // 0 .. 2*128*1024-1
  if (i >= 2 * BB * HH) return;
  int bs_last = meta[65];
  size_t base = (size_t)meta[66] * VV;  // after packed outputs
  int layer = i >> 17;                  // 128*1024 = 131072
  int rem   = i & (BB * HH - 1);
  int b = rem >> 10;
  int h = rem & (HH - 1);
  if (b >= bs_last) return;
  const float* hsrc = layer ? h1f : h0f;
  const float* csrc = layer ? c1f : c0f;
  size_t slot = ((size_t)layer * bs_last + b) * HH + h;
  out[base + slot] = hsrc[b * HH + h];
  out[base + (size_t)2 * bs_last * HH + slot] = csrc[b * HH + h];
}

// ---------------------------------------------------------------------------
extern "C" void kernel_launch(void* const* d_in, const int* in_sizes, int n_in,
                              void* d_out, int out_size, void* d_ws, size_t ws_size,
                              hipStream_t stream) {
  const float* features = (const float*)d_in[0];
  const int*   tokens   = (const int*)d_in[1];
  const int*   lengths  = (const int*)d_in[2];
  const float* embed_W  = (const float*)d_in[3];
  const float* w_ih0    = (const float*)d_in[4];
  const float* w_hh0    = (const float*)d_in[5];
  const float* b_ih0    = (const float*)d_in[6];
  const float* b_hh0    = (const float*)d_in[7];
  const float* w_ih1    = (const float*)d_in[8];
  const float* w_hh1    = (const float*)d_in[9];
  const float* b_ih1    = (const float*)d_in[10];
  const float* b_hh1    = (const float*)d_in[11];
  const float* lin_W    = (const float*)d_in[12];
  const float* lin_b    = (const float*)d_in[13];
  float* out = (float*)d_out;
  float* ws  = (float*)d_ws;

  const size_t HS = (size_t)SS * BB * HH;  // per-layer hidden sequence (8.4M floats)
  const size_t HB = (size_t)BB * HH;       // one hidden state (131072 floats)

  float* hs0    = ws;
  float* hs1    = ws + HS;
  float* h0b[2] = {ws + 2 * HS,          ws + 2 * HS + HB};
  float* h1b[2] = {ws + 2 * HS + 2 * HB, ws + 2 * HS + 3 * HB};
  float* c0     = ws + 2 * HS + 4 * HB;
  float* c1     = ws + 2 * HS + 5 * HB;
  int*   meta   = (int*)(ws + 2 * HS + 6 * HB);

  meta_kernel<<<1, 128, 0, stream>>>(lengths, meta);
  zero_kernel<<<(int)((6 * HB + 255) / 256), 256, 0, stream>>>(ws + 2 * HS,
                                                               (int)(6 * HB));

  // Run all 64 steps unconditionally; mask freezes inactive rows, so steps
  // past each sequence's length (and past T = max length) are no-ops.
  for (int t = 0; t < SS; ++t) {
    lstm_step_kernel<<<64, 128, 0, stream>>>(
        features, tokens, embed_W, /*x_seq=*/nullptr,
        h0b[t & 1], h0b[(t + 1) & 1], c0, hs0 + (size_t)t * HB,
        w_ih0, w_hh0, b_ih0, b_hh0, lengths, t, EE);
    lstm_step_kernel<<<64, 128, 0, stream>>>(
        nullptr, nullptr, nullptr, /*x_seq=*/hs0 + (size_t)t * HB,
        h1b[t & 1], h1b[(t + 1) & 1], c1, hs1 + (size_t)t * HB,
        w_ih1, w_hh1, b_ih1, b_hh1, lengths, t, HH);
  }

  proj_kernel<<<dim3(125, 256), 128, 0, stream>>>(hs1, lin_W, lin_b, lengths,
                                                  meta, out);
  // After 64 (even) steps the final h lives back in buffer 0 of each layer.
  finals_kernel<<<1024, 256, 0, stream>>>(h0b[0], h1b[0], c0, c1, meta, out);
}